// NonstationaryMaternC2Kernel_14534169329735
// MI455X (gfx1250) — compile-verified
//
#include <hip/hip_runtime.h>

typedef __attribute__((ext_vector_type(2))) float v2f;
typedef __attribute__((ext_vector_type(8))) float v8f;

#define NDIMS 3

// ---------------------------------------------------------------------------
// Kernel 1: per-point lengthscale MLP (Linear -> selu -> Linear -> softplus)
// and squared norm. Writes interleaved float2 { sqrt5 * softplus(z), ||p||^2 }
// so the pairwise kernel needs a single b64 load per point.
// ---------------------------------------------------------------------------
__global__ void scale_norm_kernel(const float* __restrict__ pts,
                                  const float* __restrict__ W1,   // [L,3]
                                  const float* __restrict__ b1,   // [L]
                                  const float* __restrict__ W2,   // [1,L]
                                  const float* __restrict__ b2,   // [1]
                                  float2* __restrict__ sn_out,    // [P] {sqrt5*s, n2}
                                  int P, int LATENT) {
    int p = blockIdx.x * blockDim.x + threadIdx.x;
    if (p >= P) return;

    float v0 = pts[p * NDIMS + 0];
    float v1 = pts[p * NDIMS + 1];
    float v2 = pts[p * NDIMS + 2];
    float n2 = v0 * v0 + v1 * v1 + v2 * v2;

    const float lam   = 1.0507009873554805f;   // jax.nn.selu scale
    const float alpha = 1.6732632423543772f;   // jax.nn.selu alpha

    float acc = b2[0];
    #pragma unroll 4
    for (int l = 0; l < LATENT; ++l) {
        float h = fmaf(W1[l * NDIMS + 0], v0,
                  fmaf(W1[l * NDIMS + 1], v1,
                  fmaf(W1[l * NDIMS + 2], v2, b1[l])));
        float neg = lam * alpha * (__expf(h) - 1.0f);
        float hs  = (h > 0.0f) ? (lam * h) : neg;
        acc = fmaf(W2[l], hs, acc);
    }
    // numerically stable softplus: max(x,0) + log1p(exp(-|x|))
    float sp = fmaxf(acc, 0.0f) + log1pf(__expf(-fabsf(acc)));
    const float sqrt5 = 2.2360679774997896f;
    sn_out[p] = make_float2(sqrt5 * sp, n2);
}

// ---------------------------------------------------------------------------
// Kernel 2: pairwise nonstationary Matern-5/2.
// One wave computes one 16x16 tile; dot products via V_WMMA_F32_16X16X4_F32
// (K = NDIMS padded to 4 with 0).
//
// WMMA f32 16x16x4 layout (ISA 7.12.2):
//   A 16x4:  VGPR0 = {lanes0-15: K=0, lanes16-31: K=2}
//            VGPR1 = {lanes0-15: K=1, lanes16-31: K=3}   (rows M = lane&15)
//   B 4x16:  same striping (cols N = lane&15)
//   D 16x16: VGPR v -> row M = v + 8*(lane>=16), col N = lane&15
//
// Matern-5/2 with t = sqrt5 * r * (sx+sy):
//   (1 + t + t^2/3) * exp(-t)            [since (5/3)rs^2 = t^2/3]
// ---------------------------------------------------------------------------
__global__ void matern_wmma_kernel(const float* __restrict__ x,     // [N,3]
                                   const float* __restrict__ y,     // [M,3]
                                   const float2* __restrict__ xsn,  // [N] {sqrt5*sx, nx2}
                                   const float2* __restrict__ ysn,  // [M] {sqrt5*sy, ny2}
                                   float* __restrict__ out,         // [N,M]
                                   int N, int M, int tiles_j) {
    const int wave = threadIdx.x >> 5;          // 8 waves per block (wave32)
    const int lane = threadIdx.x & 31;
    const int half = lane >> 4;                 // 0: lanes 0-15, 1: lanes 16-31
    const int idx  = lane & 15;

    const int tile_i = blockIdx.y;
    const int tile_j = blockIdx.x * 8 + wave;
    if (tile_j >= tiles_j) return;              // wave-uniform exit, EXEC stays full

    const int i0 = tile_i * 16;
    const int j0 = tile_j * 16;

    // clamp loads so EXEC can stay all-1s through the WMMA
    const int ai = min(i0 + idx, N - 1);
    const int bj = min(j0 + idx, M - 1);

    v2f a, b;
    a.x = x[ai * NDIMS + 2 * half];                         // K=0 or K=2
    { float t = x[ai * NDIMS + 1]; a.y = half ? 0.0f : t; } // K=1 or K=3(pad)
    b.x = y[bj * NDIMS + 2 * half];
    { float t = y[bj * NDIMS + 1]; b.y = half ? 0.0f : t; }

    v8f c = {};
    v8f d = __builtin_amdgcn_wmma_f32_16x16x4_f32(
        /*neg_a=*/false, a, /*neg_b=*/false, b,
        /*c_mod=*/(short)0, c, /*reuse_a=*/false, /*reuse_b=*/false);

    // per-lane column data (same for all 8 rows this lane owns)
    const int    col = j0 + idx;
    const float2 cy  = ysn[bj];                 // {sqrt5*sy, ny2}

    const int rowbase = i0 + half * 8;
    // 32-bit output indexing: N*M*4 = 256MB < 4GB
    unsigned int oidx = (unsigned int)rowbase * (unsigned int)M + (unsigned int)col;
    const unsigned int ostride = (unsigned int)M;

    float val[8];
    #pragma unroll
    for (int v = 0; v < 8; ++v) {
        const int rc = min(rowbase + v, N - 1);
        const float2 cx = xsn[rc];              // {sqrt5*sx, nx2} (b64, L1 broadcast)
        float r2 = fmaf(-2.0f, d[v], cx.y + cy.y);
        r2 = fmaxf(r2, 1e-12f);
        float r  = __fsqrt_rn(r2);
        float t  = r * (cx.x + cy.x);           // sqrt5 * r * (sx+sy)
        float u    = fmaf(t, 0.3333333333333333f, 1.0f);
        float poly = fmaf(t, u, 1.0f);          // 1 + t + t^2/3
        val[v] = poly * __expf(-t);
    }

    if (i0 + 16 <= N) {
        // full tile in rows: branch-free store loop; single EXEC set for cols
        if (col < M) {
            #pragma unroll
            for (int v = 0; v < 8; ++v) {
                __builtin_nontemporal_store(val[v], &out[oidx]);
                oidx += ostride;
            }
        }
    } else {
        #pragma unroll
        for (int v = 0; v < 8; ++v) {
            if (rowbase + v < N && col < M) {
                __builtin_nontemporal_store(val[v], &out[oidx]);
            }
            oidx += ostride;
        }
    }
}

// ---------------------------------------------------------------------------
// Launch
// ---------------------------------------------------------------------------
extern "C" void kernel_launch(void* const* d_in, const int* in_sizes, int n_in,
                              void* d_out, int out_size, void* d_ws, size_t ws_size,
                              hipStream_t stream) {
    const float* x  = (const float*)d_in[0];
    const float* y  = (const float*)d_in[1];
    const float* W1 = (const float*)d_in[2];
    const float* b1 = (const float*)d_in[3];
    const float* W2 = (const float*)d_in[4];
    const float* b2 = (const float*)d_in[5];

    const int N      = in_sizes[0] / NDIMS;
    const int M      = in_sizes[1] / NDIMS;
    const int LATENT = in_sizes[3];

    float2* xsn = (float2*)d_ws;        // [N]
    float2* ysn = xsn + N;              // [M]   total (N+M)*8 bytes = 128 KB

    scale_norm_kernel<<<(N + 255) / 256, 256, 0, stream>>>(x, W1, b1, W2, b2, xsn, N, LATENT);
    scale_norm_kernel<<<(M + 255) / 256, 256, 0, stream>>>(y, W1, b1, W2, b2, ysn, M, LATENT);

    const int tiles_i = (N + 15) / 16;   // 512
    const int tiles_j = (M + 15) / 16;   // 512
    dim3 grid((tiles_j + 7) / 8, tiles_i);
    matern_wmma_kernel<<<grid, dim3(256), 0, stream>>>(x, y, xsn, ysn,
                                                       (float*)d_out, N, M, tiles_j);
}